// TempoPerturbation_71708773974380
// MI455X (gfx1250) — compile-verified
//
#include <hip/hip_runtime.h>

// ---------------------------------------------------------------------------
// WSOLA tempo perturbation for gfx1250 (MI455X).
// Phase 0: build padded waveform in workspace, init delta[0]=0.
// Phase 1: single-workgroup sequential scan (8439 steps). Per step the
//          1025-lag cross-correlation is five 16x16 f32 WMMA tiles
//          (Toeplitz mapping, K~2064 split 4 ways over 20 waves).
//          Both per-step input windows are prefetched one step ahead with
//          async global->LDS copies (ASYNCcnt), double-buffered, so no
//          global-memory latency sits on the serial delta chain.
// Phase 2: deterministic gather overlap-add + Hann normalization.
// ---------------------------------------------------------------------------

typedef __attribute__((ext_vector_type(2))) float v2f;
typedef __attribute__((ext_vector_type(8))) float v8f;

#define N_WIN     2048
#define HS        1024
#define DMAX      512
#define WAV_LEN   9600000
#define NEW_LEN   8640000          // ceil(0.9 * WAV_LEN)
#define F_FRAMES  8439             // len(arange(0, NEW_LEN + 1024, 1024))
#define LEFT_PAD  1536             // N/2 + DMAX
#define RIGHT_PAD 4608             // ceil(1/min_frac)=2 -> 2*N + DMAX
#define PAD_LEN   (LEFT_PAD + WAV_LEN + RIGHT_PAD)   // 9,606,144

#define SCAN_THREADS 640           // 20 waves (wave32)
#define RNG_N        3328          // covers l0<=1024, 16m<=240, k<=2063
#define RNG_SK       3536          // skewed: i + i/16  (bank-conflict-free A reads)
#define HSUP_N       3072          // template superset (independent of delta)
#define HBUF         2080          // 16 leading zeros + 2048 taps + 16 tail zeros

// ha_b[i] = round(1024*i / 0.9) + DMAX = round(10240*i/9) + 512
// (fraction is never exactly .5, so floor(x+0.5) == np.round)
__device__ __forceinline__ int ha_of(int i) {
    long long v = (20480LL * (long long)i + 9LL) / 18LL;
    return (int)v + DMAX;
}

// One per-lane async global->LDS dword copy (gfx1250, tracked by ASYNCcnt).
__device__ __forceinline__ void issue_async_f32(const float* g, float* lds_ptr) {
    unsigned long long ga = (unsigned long long)(uintptr_t)g;
    unsigned lo = (unsigned)(uintptr_t)lds_ptr;   // low 32 bits = LDS offset
    asm volatile("global_load_async_to_lds_b32 %0, %1, off"
                 :: "v"(lo), "v"(ga) : "memory");
}
__device__ __forceinline__ void wait_async0() {
    asm volatile("s_wait_asynccnt 0x0" ::: "memory");
}

// ---------------------------------------------------------------------------
__global__ void wsola_pad_kernel(const float* __restrict__ wav,
                                 float* __restrict__ wav_pad,
                                 int* __restrict__ delta) {
    long long i = (long long)blockIdx.x * 256LL + threadIdx.x;
    if (i < PAD_LEN) {
        long long j = i - LEFT_PAD;
        wav_pad[i] = (j >= 0 && j < WAV_LEN) ? wav[j] : 0.0f;
    }
    if (i == 0) delta[0] = 0;
}

// ---------------------------------------------------------------------------
__global__ __launch_bounds__(SCAN_THREADS, 1)
void wsola_scan_kernel(const float* __restrict__ wav_pad,
                       int* __restrict__ delta) {
    __shared__ float rngb[2][RNG_SK];        // search windows (skewed layout)
    __shared__ float hsup[2][HSUP_N];        // template supersets
    __shared__ float hb[HBUF];               // zero-bordered h_pad for B matrix
    __shared__ float part[20 * 32 * 8];      // per-wave partial D tiles
    __shared__ float bestv[20];
    __shared__ int   bestl[20];
    __shared__ int   s_delta;

    const int tid    = threadIdx.x;
    const int wave   = tid >> 5;
    const int lane   = tid & 31;
    const int tile   = wave >> 2;            // 0..4  -> lag base l0 = 256*tile
    const int kslice = wave & 3;             // 0..3  -> K partition
    const int m      = lane & 15;            // A row (and B column n)
    const int koff   = (lane >> 4) << 1;     // 0 or 2: k-subindex per ISA layout
    const int l0     = tile << 8;

    if (tid < 16)              hb[tid] = 0.0f;              // h_pad[k<0]    = 0
    if (tid >= 16 && tid < 32) hb[2064 + (tid - 16)] = 0.0f; // h_pad[k>=2048]=0
    if (tid == 0)              s_delta = 0;

    // Prologue: prefetch buffers for step 0 (neither depends on delta).
    {
        const int rbase = ha_of(1) - DMAX;
        for (int i = tid; i < RNG_N; i += SCAN_THREADS)
            issue_async_f32(wav_pad + rbase + i, &rngb[0][i + (i >> 4)]);
        const int sbase = ha_of(0) + DMAX;
        for (int i = tid; i < HSUP_N; i += SCAN_THREADS)
            issue_async_f32(wav_pad + sbase + i, &hsup[0][i]);
    }
    wait_async0();
    __syncthreads();

    for (int t = 0; t < F_FRAMES; ++t) {
        const int cur = t & 1, nxt = cur ^ 1;

        // Kick off next step's windows while this step computes.
        if (t + 1 < F_FRAMES) {
            const int tn = t + 1;
            const int rbase = ha_of(tn + 1 < F_FRAMES ? tn + 1 : F_FRAMES - 1) - DMAX;
            for (int i = tid; i < RNG_N; i += SCAN_THREADS)
                issue_async_f32(wav_pad + rbase + i, &rngb[nxt][i + (i >> 4)]);
            const int sbase = ha_of(tn) + DMAX;
            for (int i = tid; i < HSUP_N; i += SCAN_THREADS)
                issue_async_f32(wav_pad + sbase + i, &hsup[nxt][i]);
        }

        // Re-slice template at current delta (tiny LDS->LDS copy).
        const int hoff = s_delta + DMAX;     // 0..1024
        for (int n = tid; n < N_WIN; n += SCAN_THREADS)
            hb[16 + n] = hsup[cur][hoff + n];
        __syncthreads();

        // corr[l0 + 16m + n] = sum_k rng[l0+16m+k] * h_pad[k-n]
        const float* rng = rngb[cur];
        v8f c = {};
        const int kk0 = kslice * 129;        // 516 K-steps of 4, split 4 ways
        #pragma unroll 4
        for (int j = 0; j < 129; ++j) {
            const int kb = (kk0 + j) << 2;
            v2f a, b;
            const int ar  = l0 + (m << 4) + kb + koff;   // ar%16 != 15
            const int sar = ar + (ar >> 4);              // skewed index
            a.x = rng[sar];
            a.y = rng[sar + 1];
            const int br = 16 + kb + koff - m;           // B[k,n]=h_pad[k-n], n==m
            b.x = hb[br];
            b.y = hb[br + 1];
            c = __builtin_amdgcn_wmma_f32_16x16x4_f32(
                    false, a, false, b, (short)0, c, false, false);
        }

        float* pp = &part[(wave << 8) + (lane << 3)];
        #pragma unroll
        for (int r = 0; r < 8; ++r) pp[r] = c[r];
        __syncthreads();

        // Reduce K-slices, map D elements to lags, local argmax.
        float bv = -3.4e38f;
        int   bl = -1;
        for (int idx = tid; idx < 1280; idx += SCAN_THREADS) {
            const int tl  = idx >> 8;        // tile
            const int rem = idx & 255;
            const int ln  = rem >> 3;        // lane within tile
            const int r   = rem & 7;         // C/D VGPR index
            float v = 0.0f;
            #pragma unroll
            for (int s = 0; s < 4; ++s)
                v += part[(((tl << 2) + s) << 8) + (ln << 3) + r];
            // D layout: M = r + 8*(lane>=16), N = lane&15 ; lag = l0 + 16*M + N
            const int lag = (tl << 8) + ((r + ((ln >> 4) << 3)) << 4) + (ln & 15);
            if (lag <= 2 * DMAX) {
                // np.argmax on reversed corr -> ties resolve to LARGER lag
                if (v > bv || (v == bv && lag > bl)) { bv = v; bl = lag; }
            }
        }
        // wave32 butterfly argmax (no barriers)
        #pragma unroll
        for (int off = 16; off; off >>= 1) {
            const float ov = __shfl_xor(bv, off, 32);
            const int   ol = __shfl_xor(bl, off, 32);
            if (ov > bv || (ov == bv && ol > bl)) { bv = ov; bl = ol; }
        }
        if (lane == 0) { bestv[wave] = bv; bestl[wave] = bl; }
        __syncthreads();

        if (wave == 0) {
            float fv = (lane < 20) ? bestv[lane] : -3.4e38f;
            int   fl = (lane < 20) ? bestl[lane] : -1;
            #pragma unroll
            for (int off = 16; off; off >>= 1) {
                const float ov = __shfl_xor(fv, off, 32);
                const int   ol = __shfl_xor(fl, off, 32);
                if (ov > fv || (ov == fv && ol > fl)) { fv = ov; fl = ol; }
            }
            if (lane == 0) {
                const int dn = fl - DMAX;    // delta_next = lag_best - DMAX
                delta[t + 1] = dn;
                s_delta = dn;
            }
        }
        // Next step's LDS buffers must be fully written and visible.
        wait_async0();
        __syncthreads();
    }
}

// ---------------------------------------------------------------------------
// Each output sample x = i + N/2 is covered by exactly <=2 frames
// (hop = N/2), so gather deterministically instead of atomic scatter.
__global__ void wsola_ola_kernel(const float* __restrict__ wav_pad,
                                 const int* __restrict__ delta,
                                 float* __restrict__ out) {
    long long i = (long long)blockIdx.x * 256LL + threadIdx.x;
    if (i >= NEW_LEN) return;
    const long long x  = i + HS;
    const int t1 = (int)(x >> 10);
    float acc = 0.0f, wacc = 0.0f;
    #pragma unroll
    for (int d = 0; d < 2; ++d) {
        const int t = t1 - d;
        if (t >= 0 && t < F_FRAMES) {
            const int n = (int)(x - ((long long)t << 10));
            if (n >= 0 && n < N_WIN) {
                const float w = 0.5f *
                    (1.0f - __cosf(3.06796157577128246e-3f * (float)n)); // 2*pi/2048
                const int st = ha_of(t) + delta[t] + n;
                acc  += wav_pad[st] * w;
                wacc += w;
            }
        }
    }
    if (wacc < 0.001f) wacc = 1.0f;
    out[i] = acc / wacc;
}

// ---------------------------------------------------------------------------
extern "C" void kernel_launch(void* const* d_in, const int* in_sizes, int n_in,
                              void* d_out, int out_size, void* d_ws, size_t ws_size,
                              hipStream_t stream) {
    (void)in_sizes; (void)n_in; (void)out_size; (void)ws_size;
    const float* wav = (const float*)d_in[0];
    float* wav_pad   = (float*)d_ws;
    int*   delta     = (int*)((char*)d_ws + (size_t)PAD_LEN * sizeof(float));
    float* out       = (float*)d_out;

    const int padBlocks = (PAD_LEN + 255) / 256;
    wsola_pad_kernel<<<padBlocks, 256, 0, stream>>>(wav, wav_pad, delta);

    wsola_scan_kernel<<<1, SCAN_THREADS, 0, stream>>>(wav_pad, delta);

    const int olaBlocks = (NEW_LEN + 255) / 256;
    wsola_ola_kernel<<<olaBlocks, 256, 0, stream>>>(wav_pad, delta, out);
}